// YoloLayer_26233660244542
// MI455X (gfx1250) — compile-verified
//
#include <hip/hip_runtime.h>
#include <stdint.h>

// Problem constants (match reference)
constexpr int B_  = 16;
constexpr int A_  = 3;
constexpr int C_  = 80;
constexpr int H_  = 76;
constexpr int W_  = 76;
constexpr int HW_ = H_ * W_;        // 5776
constexpr int CHN = 5 + C_;         // 85
constexpr int ROWS  = 128;          // spatial rows per block
constexpr int TILES = (HW_ + ROWS - 1) / ROWS;   // 46
constexpr int ISTR  = ROWS + 4;     // 132 floats: 528B channel rows (16B aligned),
                                    // bank stride 4 -> conflict-free column access

__device__ __forceinline__ float fast_exp(float x) {
    return __builtin_amdgcn_exp2f(x * 1.44269504088896340736f);   // v_exp_f32
}
__device__ __forceinline__ float fast_sigmoid(float x) {
    return __builtin_amdgcn_rcpf(1.0f + fast_exp(-x));            // v_rcp_f32
}

// LDS byte offset of a generic pointer to a __shared__ object:
// generic LDS addr = {shared_aperture, offset}; HW truncates addr[31:0].
__device__ __forceinline__ uint32_t lds_offset(const void* p) {
    return (uint32_t)(uintptr_t)p;
}

__global__ __launch_bounds__(ROWS) void yolo_layer_kernel(
    const float* __restrict__ in,     // [B, A, 85, H, W]
    float*       __restrict__ boxes,  // [B*N, 4]  (row = ba*HW + hw)
    float*       __restrict__ confs)  // [B*N, C]
{
    __shared__ __align__(16) float smem[C_ * ISTR]; // 42,240B class tile, in-place softmax
    __shared__ float scale_l[ROWS];                 // per-row det_conf / sum

    const int tile = blockIdx.x % TILES;
    const int ba   = blockIdx.x / TILES;      // b*A + a
    const int a    = ba % A_;
    const int hw0  = tile * ROWS;
    const int r    = threadIdx.x;
    const int hw   = hw0 + r;

    const float* base = in + (size_t)ba * CHN * HW_;

    // ---- stage 80x128 class tile into LDS via async DMA (b128 per lane) ----
    // Thread r owns hw-chunk j4 = (r&31)*4 of channels c = (r>>5) + 4*iter.
    // Per-iteration address steps are constants: +4*HW*4 global, +4*ISTR*4 LDS.
    {
        const int j4 = (r & 31) * 4;            // fixed hw offset of this thread's chunk
        const int c0 = r >> 5;                  // starting channel: 0..3
        if (hw0 + j4 < HW_) {                   // loop-invariant (HW%4==0: chunks never partial)
            uint32_t lb = lds_offset(&smem[c0 * ISTR + j4]);
            unsigned long long ga = (unsigned long long)(uintptr_t)
                (base + (size_t)(5 + c0) * HW_ + hw0 + j4);
            #pragma unroll
            for (int it = 0; it < C_ / 4; ++it) {   // 20 async b128 transfers
                asm volatile("global_load_async_to_lds_b128 %0, %1, off"
                             :: "v"(lb), "v"(ga) : "memory");
                lb += 4u * ISTR * 4u;                        // 2,112 B
                ga += (unsigned long long)(4 * HW_ * 4);     // 92,416 B
            }
        }
    }

    // ---- box channels: plain coalesced loads (overlap with async DMA) ----
    float tx = 0.f, ty = 0.f, tw = 0.f, th = 0.f, obj = 0.f;
    if (hw < HW_) {
        tx  = base[0 * HW_ + hw];
        ty  = base[1 * HW_ + hw];
        tw  = base[2 * HW_ + hw];
        th  = base[3 * HW_ + hw];
        obj = base[4 * HW_ + hw];
    }

    asm volatile("s_wait_asynccnt 0x0" ::: "memory");
    __syncthreads();

    if (hw < HW_) {
        // ---------------- boxes ----------------
        const float gx = (float)(hw % W_);
        const float gy = (float)(hw / W_);
        const float aw = (a == 0) ? 1.5f : ((a == 1) ? 2.375f : 5.0f);
        const float ah = (a == 0) ? 2.0f : ((a == 1) ? 4.5f  : 3.5f);

        const float bx = (fast_sigmoid(tx) * 1.2f - 0.1f + gx) * (1.0f / W_);
        const float by = (fast_sigmoid(ty) * 1.2f - 0.1f + gy) * (1.0f / H_);
        const float bw = fast_exp(tw) * (aw * (1.0f / W_));
        const float bh = fast_exp(th) * (ah * (1.0f / H_));
        const float bx1 = bx - 0.5f * bw;
        const float by1 = by - 0.5f * bh;
        float4 box;
        box.x = bx1; box.y = by1; box.z = bx1 + bw; box.w = by1 + bh;
        reinterpret_cast<float4*>(boxes)[(size_t)ba * HW_ + hw] = box;

        // ------ softmax over this thread's LDS column (in place) ------
        float m = -3.0e38f;
        #pragma unroll
        for (int c = 0; c < C_; ++c) m = fmaxf(m, smem[c * ISTR + r]);
        float s = 0.0f;
        #pragma unroll
        for (int c = 0; c < C_; ++c) {
            const float e = fast_exp(smem[c * ISTR + r] - m);
            smem[c * ISTR + r] = e;      // column r owned by this thread only
            s += e;
        }
        scale_l[r] = fast_sigmoid(obj) * __builtin_amdgcn_rcpf(s);
    }
    __syncthreads();

    // ---- contiguous b128 write-out of this block's conf region ----
    const int nvalid = min(ROWS, HW_ - hw0);
    float4* ob4 = reinterpret_cast<float4*>(confs + (size_t)(ba * HW_ + hw0) * C_);
    const int total4 = nvalid * (C_ / 4);
    for (int i = threadIdx.x; i < total4; i += ROWS) {
        const int e4 = i * 4;
        const int rr = e4 / C_;
        const int cc = e4 - rr * C_;
        const float sc = scale_l[rr];
        float4 o;
        o.x = smem[(cc + 0) * ISTR + rr] * sc;
        o.y = smem[(cc + 1) * ISTR + rr] * sc;
        o.z = smem[(cc + 2) * ISTR + rr] * sc;
        o.w = smem[(cc + 3) * ISTR + rr] * sc;
        ob4[i] = o;
    }
}

extern "C" void kernel_launch(void* const* d_in, const int* in_sizes, int n_in,
                              void* d_out, int out_size, void* d_ws, size_t ws_size,
                              hipStream_t stream) {
    (void)in_sizes; (void)n_in; (void)d_ws; (void)ws_size; (void)out_size;
    const float* in = (const float*)d_in[0];
    float* boxes = (float*)d_out;                                  // B*N*4 floats
    float* confs = (float*)d_out + (size_t)B_ * A_ * HW_ * 4;      // then B*N*C
    const int grid = B_ * A_ * TILES;   // 16*3*46 = 2208 blocks
    yolo_layer_kernel<<<grid, ROWS, 0, stream>>>(in, boxes, confs);
}